// StockMambaL2_14748917695060
// MI455X (gfx1250) — compile-verified
//
#include <hip/hip_runtime.h>
#include <math.h>

#define B_   4
#define T_   1024
#define D_   512
#define L_   4
#define N_   16
#define DC_  4
#define DI_  1024
#define DTR_ 32
#define H_   128
#define BT_  (B_*T_)
#define EPS_ 1e-5f

typedef float v2f __attribute__((ext_vector_type(2)));
typedef float v8f __attribute__((ext_vector_type(8)));

#if __has_builtin(__builtin_amdgcn_global_load_async_to_lds_b128) && \
    __has_builtin(__builtin_amdgcn_s_wait_asynccnt)
#define USE_ASYNC_LDS 1
// exact builtin parameter types: int4 (GCC vector) in AS1 (global) / AS3 (LDS)
typedef int v4i __attribute__((vector_size(16)));
typedef __attribute__((address_space(1))) v4i GV4;
typedef __attribute__((address_space(3))) v4i LV4;
#else
#define USE_ASYNC_LDS 0
#endif

// ---------------------------------------------------------------------------
// WMMA fp32 GEMM: C[M,Nn] = act(A[M,K] (row stride lda) * W[Nn,K]^T + bias)
// Block = 256 threads = 8 waves (2x4 wave grid) -> 32x64 output tile.
// K staged through LDS in chunks of 32 via b128 transfers; each wave issues
// 8 back-to-back v_wmma per chunk (fragments preloaded into registers).
// Staging uses gfx1250 async-to-LDS DMA (ASYNCcnt) when the toolchain
// exposes it; otherwise a register-pipelined b128 path.
// act: 0 = none, 1 = softplus (for dt_proj).
// ---------------------------------------------------------------------------
__global__ __launch_bounds__(256) void k_gemm(const float* __restrict__ A, int lda,
    const float* __restrict__ W, const float* __restrict__ bias,
    float* __restrict__ C, int M, int Nn, int K, int act)
{
  // padded rows (36 floats = 144B): b128-aligned stores, conflict-free reads
  __shared__ float As[32][36];   // As[m][k]
  __shared__ float Bs[64][36];   // Bs[n][k]

  const int tid  = threadIdx.x;
  const int wave = tid >> 5, lane = tid & 31;
  const int wm = wave >> 2, wn = wave & 3;
  const int m0 = blockIdx.y * 32, n0 = blockIdx.x * 64;

  v8f c = {0.f, 0.f, 0.f, 0.f, 0.f, 0.f, 0.f, 0.f};

  // A frag (16x4 f32): lanes 0-15 hold K=0,1 ; lanes 16-31 hold K=2,3 (ISA 7.12.2)
  // B frag (4x16 f32): v0 = rows K=0/K=2 striped across lane halves, v1 = K=1/K=3
  const int mrow = wm * 16 + (lane & 15);
  const int ncol = wn * 16 + (lane & 15);
  const int kh   = (lane >> 4) * 2;

  // staging slots: one float4 of A + two float4 of W per thread per chunk
  const int am = tid >> 3, ak = (tid & 7) * 4;           // 32 x 8 float4
  const int bn0 = tid >> 3, bk0 = (tid & 7) * 4;         // first 256 slots
  const int bn1 = (tid + 256) >> 3, bk1 = (tid & 7) * 4; // next 256 slots

  const float* gA  = A + (size_t)(m0 + am) * lda + ak;
  const float* gB0 = W + (size_t)(n0 + bn0) * K + bk0;
  const float* gB1 = W + (size_t)(n0 + bn1) * K + bk1;

#if USE_ASYNC_LDS
  // -------- async DMA staging (GLOBAL_LOAD_ASYNC_TO_LDS_B128, ASYNCcnt) ----
  for (int k0 = 0; k0 < K; k0 += 32) {
    __builtin_amdgcn_global_load_async_to_lds_b128(
        (GV4*)(gA + k0),  (LV4*)&As[am][ak],   0, 0);
    __builtin_amdgcn_global_load_async_to_lds_b128(
        (GV4*)(gB0 + k0), (LV4*)&Bs[bn0][bk0], 0, 0);
    __builtin_amdgcn_global_load_async_to_lds_b128(
        (GV4*)(gB1 + k0), (LV4*)&Bs[bn1][bk1], 0, 0);
    __builtin_amdgcn_s_wait_asynccnt(0);
    __syncthreads();

    v2f afr[8], bfr[8];
    #pragma unroll
    for (int q = 0; q < 8; ++q) {
      int kk = q * 4;
      afr[q].x = As[mrow][kk + kh]; afr[q].y = As[mrow][kk + kh + 1];
      bfr[q].x = Bs[ncol][kk + kh]; bfr[q].y = Bs[ncol][kk + kh + 1];
    }
    #pragma unroll
    for (int q = 0; q < 8; ++q)
      c = __builtin_amdgcn_wmma_f32_16x16x4_f32(false, afr[q], false, bfr[q],
                                                (short)0, c, false, false);
    __syncthreads();
  }
#else
  // -------- register-pipelined staging (b128 loads, b128 LDS stores) -------
  float4 aR, bR0, bR1;
  aR  = *(const float4*)(gA);
  bR0 = *(const float4*)(gB0);
  bR1 = *(const float4*)(gB1);

  for (int k0 = 0; k0 < K; k0 += 32) {
    *(float4*)&As[am][ak]   = aR;
    *(float4*)&Bs[bn0][bk0] = bR0;
    *(float4*)&Bs[bn1][bk1] = bR1;
    __syncthreads();

    if (k0 + 32 < K) {                 // prefetch next chunk during compute
      aR  = *(const float4*)(gA  + k0 + 32);
      bR0 = *(const float4*)(gB0 + k0 + 32);
      bR1 = *(const float4*)(gB1 + k0 + 32);
    }

    v2f afr[8], bfr[8];
    #pragma unroll
    for (int q = 0; q < 8; ++q) {
      int kk = q * 4;
      afr[q].x = As[mrow][kk + kh]; afr[q].y = As[mrow][kk + kh + 1];
      bfr[q].x = Bs[ncol][kk + kh]; bfr[q].y = Bs[ncol][kk + kh + 1];
    }
    #pragma unroll
    for (int q = 0; q < 8; ++q)
      c = __builtin_amdgcn_wmma_f32_16x16x4_f32(false, afr[q], false, bfr[q],
                                                (short)0, c, false, false);
    __syncthreads();
  }
#endif

  // C layout: VGPR v -> M = v (lanes 0-15) / v+8 (lanes 16-31), N = lane%16
  const int gn  = n0 + ncol;
  const int gm0 = m0 + wm * 16 + (lane >> 4) * 8;
  float bv = bias ? bias[gn] : 0.f;
  #pragma unroll
  for (int v = 0; v < 8; ++v) {
    float val = c[v] + bv;
    if (act == 1) val = (val > 20.f) ? val : log1pf(__expf(val));  // softplus
    C[(size_t)(gm0 + v) * Nn + gn] = val;
  }
}

// ---------------------------------------------------------------------------
// Depthwise causal conv (DC=4) over T + bias + SiLU. Input = first DI channels
// of xz [B,T,2*DI]; output xc [B,T,DI].
// ---------------------------------------------------------------------------
__global__ __launch_bounds__(256) void k_conv_silu(const float* __restrict__ xz,
    const float* __restrict__ cw, const float* __restrict__ cb,
    float* __restrict__ xc)
{
  int idx = blockIdx.x * 256 + threadIdx.x;           // (b*T+t)*DI + d
  int d  = idx & (DI_ - 1);
  int bt = idx / DI_;
  int t  = bt & (T_ - 1);
  float acc = cb[d];
  #pragma unroll
  for (int j = 0; j < DC_; ++j) {
    int tt = t - (DC_ - 1) + j;
    if (tt >= 0)
      acc += xz[(size_t)(bt - (DC_ - 1) + j) * (2 * DI_) + d] * cw[d * DC_ + j];
  }
  xc[idx] = acc / (1.f + __expf(-acc));               // silu
}

// ---------------------------------------------------------------------------
// Selective scan: one thread per (b,d); 16 states in registers.
// Fuses D-skip and SiLU(z) gating. Bm/Cm loads are wave-uniform (same b,t).
// ---------------------------------------------------------------------------
__global__ __launch_bounds__(256) void k_scan(const float* __restrict__ dt,
    const float* __restrict__ xc, const float* __restrict__ dbc,
    const float* __restrict__ xz, const float* __restrict__ A_log,
    const float* __restrict__ D_skip, float* __restrict__ y)
{
  int tid = blockIdx.x * 256 + threadIdx.x;           // b*DI + d
  int d = tid & (DI_ - 1);
  int b = tid / DI_;
  float A[N_], h[N_];
  #pragma unroll
  for (int n = 0; n < N_; ++n) { A[n] = -__expf(A_log[d * N_ + n]); h[n] = 0.f; }
  float dsk = D_skip[d];

  for (int t = 0; t < T_; ++t) {
    size_t bt = (size_t)b * T_ + t;
    float dtv = dt[bt * DI_ + d];
    float xv  = xc[bt * DI_ + d];
    float zv  = xz[bt * (2 * DI_) + DI_ + d];
    const float* BC = dbc + bt * (DTR_ + 2 * N_);
    float acc = 0.f;
    #pragma unroll
    for (int n = 0; n < N_; ++n) {
      float dA = __expf(dtv * A[n]);
      h[n] = dA * h[n] + dtv * BC[DTR_ + n] * xv;     // dBx
      acc += h[n] * BC[DTR_ + N_ + n];                // * C
    }
    float g = zv / (1.f + __expf(-zv));               // silu(z)
    y[bt * DI_ + d] = (acc + xv * dsk) * g;
  }
}

// ---------------------------------------------------------------------------
// LayerNorm(row of D=512) * g + b + residual. One block per (b,t) row.
// ---------------------------------------------------------------------------
__global__ __launch_bounds__(256) void k_ln_res(const float* __restrict__ yp,
    const float* __restrict__ xprev, const float* __restrict__ g,
    const float* __restrict__ bb, float* __restrict__ xout)
{
  __shared__ float red[256];
  const int row = blockIdx.x, tid = threadIdx.x;
  const float* yr = yp + (size_t)row * D_;
  float v0 = yr[tid], v1 = yr[tid + 256];

  red[tid] = v0 + v1; __syncthreads();
  for (int s = 128; s > 0; s >>= 1) { if (tid < s) red[tid] += red[tid + s]; __syncthreads(); }
  float mu = red[0] / D_; __syncthreads();

  float d0 = v0 - mu, d1 = v1 - mu;
  red[tid] = d0 * d0 + d1 * d1; __syncthreads();
  for (int s = 128; s > 0; s >>= 1) { if (tid < s) red[tid] += red[tid + s]; __syncthreads(); }
  float rs = rsqrtf(red[0] / D_ + EPS_);

  const float* xr = xprev + (size_t)row * D_;
  xout[(size_t)row * D_ + tid]       = d0 * rs * g[tid]       + bb[tid]       + xr[tid];
  xout[(size_t)row * D_ + tid + 256] = d1 * rs * g[tid + 256] + bb[tid + 256] + xr[tid + 256];
}

// Attention scores: one thread per (b,t).
__global__ __launch_bounds__(256) void k_scores(const float* __restrict__ x,
    const float* __restrict__ aw, const float* __restrict__ ab,
    float* __restrict__ sc)
{
  int bt = blockIdx.x * 256 + threadIdx.x;
  float s = ab[0];
  const float* xr = x + (size_t)bt * D_;
  for (int d2 = 0; d2 < D_; ++d2) s += xr[d2] * aw[d2];
  sc[bt] = s;
}

// Softmax over T + weighted pooling. One block per b; probs cached in LDS.
__global__ __launch_bounds__(256) void k_pool(const float* __restrict__ x,
    const float* __restrict__ sc, float* __restrict__ pooled)
{
  __shared__ float red[256];
  __shared__ float sp[T_];
  const int b = blockIdx.x, tid = threadIdx.x;
  const float* s = sc + b * T_;

  float mx = -3.4e38f;
  for (int t = tid; t < T_; t += 256) mx = fmaxf(mx, s[t]);
  red[tid] = mx; __syncthreads();
  for (int st = 128; st > 0; st >>= 1) { if (tid < st) red[tid] = fmaxf(red[tid], red[tid + st]); __syncthreads(); }
  mx = red[0]; __syncthreads();

  float sum = 0.f;
  for (int t = tid; t < T_; t += 256) { float e = expf(s[t] - mx); sp[t] = e; sum += e; }
  red[tid] = sum; __syncthreads();
  for (int st = 128; st > 0; st >>= 1) { if (tid < st) red[tid] += red[tid + st]; __syncthreads(); }
  float inv = 1.f / red[0]; __syncthreads();

  for (int d2 = tid; d2 < D_; d2 += 256) {
    float acc = 0.f;
    for (int t = 0; t < T_; ++t)
      acc += sp[t] * x[((size_t)b * T_ + t) * D_ + d2];
    pooled[b * D_ + d2] = acc * inv;
  }
}

// MLP head: gelu(LN(p@W1^T+b1)) -> gelu(LN(.@W2^T+b2)) -> @W3^T+b3. Block per b.
__global__ __launch_bounds__(128) void k_head(const float* __restrict__ pooled,
    const float* __restrict__ w1, const float* __restrict__ b1,
    const float* __restrict__ g1, const float* __restrict__ bb1,
    const float* __restrict__ w2, const float* __restrict__ b2,
    const float* __restrict__ g2, const float* __restrict__ bb2,
    const float* __restrict__ w3, const float* __restrict__ b3,
    float* __restrict__ out)
{
  __shared__ float buf[H_];
  __shared__ float red[H_];
  const int b = blockIdx.x, h = threadIdx.x;
  const float* p = pooled + b * D_;

  float v = b1[h];
  for (int k = 0; k < D_; ++k) v += p[k] * w1[h * D_ + k];

  red[h] = v; __syncthreads();
  for (int s = 64; s > 0; s >>= 1) { if (h < s) red[h] += red[h + s]; __syncthreads(); }
  float mu = red[0] / H_; __syncthreads();
  float dv = v - mu;
  red[h] = dv * dv; __syncthreads();
  for (int s = 64; s > 0; s >>= 1) { if (h < s) red[h] += red[h + s]; __syncthreads(); }
  v = dv * rsqrtf(red[0] / H_ + EPS_) * g1[h] + bb1[h];
  v = 0.5f * v * (1.f + erff(v * 0.70710678118f));    // exact gelu
  buf[h] = v; __syncthreads();

  float v2 = b2[h];
  for (int k = 0; k < H_; ++k) v2 += buf[k] * w2[h * H_ + k];
  __syncthreads();
  red[h] = v2; __syncthreads();
  for (int s = 64; s > 0; s >>= 1) { if (h < s) red[h] += red[h + s]; __syncthreads(); }
  mu = red[0] / H_; __syncthreads();
  dv = v2 - mu;
  red[h] = dv * dv; __syncthreads();
  for (int s = 64; s > 0; s >>= 1) { if (h < s) red[h] += red[h + s]; __syncthreads(); }
  v2 = dv * rsqrtf(red[0] / H_ + EPS_) * g2[h] + bb2[h];
  v2 = 0.5f * v2 * (1.f + erff(v2 * 0.70710678118f));
  buf[h] = v2 * w3[h]; __syncthreads();

  red[h] = buf[h]; __syncthreads();
  for (int s = 64; s > 0; s >>= 1) { if (h < s) red[h] += red[h + s]; __syncthreads(); }
  if (h == 0) out[b] = red[0] + b3[0];
}

// ---------------------------------------------------------------------------
extern "C" void kernel_launch(void* const* d_in, const int* in_sizes, int n_in,
                              void* d_out, int out_size, void* d_ws, size_t ws_size,
                              hipStream_t stream) {
  (void)in_sizes; (void)n_in; (void)out_size; (void)ws_size;
  const float* x_in   = (const float*)d_in[0];
  const float* in_w   = (const float*)d_in[1];
  const float* conv_w = (const float*)d_in[2];
  const float* conv_b = (const float*)d_in[3];
  const float* xp_w   = (const float*)d_in[4];
  const float* dt_w   = (const float*)d_in[5];
  const float* dt_b   = (const float*)d_in[6];
  const float* A_log  = (const float*)d_in[7];
  const float* D_skip = (const float*)d_in[8];
  const float* out_w  = (const float*)d_in[9];
  const float* ln_g   = (const float*)d_in[10];
  const float* ln_b   = (const float*)d_in[11];
  const float* attn_w = (const float*)d_in[12];
  const float* attn_b = (const float*)d_in[13];
  const float* h1_w   = (const float*)d_in[14];
  const float* h1_b   = (const float*)d_in[15];
  const float* ln1_g  = (const float*)d_in[16];
  const float* ln1_b  = (const float*)d_in[17];
  const float* h2_w   = (const float*)d_in[18];
  const float* h2_b   = (const float*)d_in[19];
  const float* ln2_g  = (const float*)d_in[20];
  const float* ln2_b  = (const float*)d_in[21];
  const float* h3_w   = (const float*)d_in[22];
  const float* h3_b   = (const float*)d_in[23];
  float* out = (float*)d_out;

  // workspace carve-up (~110 MB of fp32)
  float* ws = (float*)d_ws;
  size_t o = 0;
  float* xz     = ws + o; o += (size_t)BT_ * 2 * DI_;   // [4096,2048]
  float* xc     = ws + o; o += (size_t)BT_ * DI_;       // [4096,1024]
  float* dbc    = ws + o; o += (size_t)BT_ * 64;        // [4096,64]
  float* dt     = ws + o; o += (size_t)BT_ * DI_;
  float* yb     = ws + o; o += (size_t)BT_ * DI_;
  float* yproj  = ws + o; o += (size_t)BT_ * D_;
  float* xA     = ws + o; o += (size_t)BT_ * D_;
  float* xB     = ws + o; o += (size_t)BT_ * D_;
  float* scores = ws + o; o += (size_t)BT_;
  float* pooled = ws + o; o += (size_t)B_ * D_;

  const float* xcur = x_in;
  float* bufs[2] = { xA, xB };

  for (int l = 0; l < L_; ++l) {
    // in_proj: [4096,512] x [2048,512]^T -> xz [4096,2048]
    k_gemm<<<dim3(2 * DI_ / 64, BT_ / 32), 256, 0, stream>>>(
        xcur, D_, in_w + (size_t)l * 2 * DI_ * D_, nullptr, xz, BT_, 2 * DI_, D_, 0);
    // conv + silu -> xc
    k_conv_silu<<<(BT_ * DI_) / 256, 256, 0, stream>>>(
        xz, conv_w + (size_t)l * DI_ * DC_, conv_b + (size_t)l * DI_, xc);
    // x_proj: [4096,1024] x [64,1024]^T -> dbc [4096,64]
    k_gemm<<<dim3(64 / 64, BT_ / 32), 256, 0, stream>>>(
        xc, DI_, xp_w + (size_t)l * 64 * DI_, nullptr, dbc, BT_, 64, DI_, 0);
    // dt_proj (A = dbc[:, :32], lda=64) + bias + softplus -> dt [4096,1024]
    k_gemm<<<dim3(DI_ / 64, BT_ / 32), 256, 0, stream>>>(
        dbc, 64, dt_w + (size_t)l * DI_ * DTR_, dt_b + (size_t)l * DI_,
        dt, BT_, DI_, DTR_, 1);
    // selective scan (fused D-skip + silu(z) gate) -> yb
    k_scan<<<(B_ * DI_) / 256, 256, 0, stream>>>(
        dt, xc, dbc, xz, A_log + (size_t)l * DI_ * N_, D_skip + (size_t)l * DI_, yb);
    // out_proj: [4096,1024] x [512,1024]^T -> yproj [4096,512]
    k_gemm<<<dim3(D_ / 64, BT_ / 32), 256, 0, stream>>>(
        yb, DI_, out_w + (size_t)l * D_ * DI_, nullptr, yproj, BT_, D_, DI_, 0);
    // layernorm + residual -> next x
    float* xnxt = bufs[l & 1];
    k_ln_res<<<BT_, 256, 0, stream>>>(
        yproj, xcur, ln_g + (size_t)l * D_, ln_b + (size_t)l * D_, xnxt);
    xcur = xnxt;
  }

  k_scores<<<BT_ / 256, 256, 0, stream>>>(xcur, attn_w, attn_b, scores);
  k_pool<<<B_, 256, 0, stream>>>(xcur, scores, pooled);
  k_head<<<B_, 128, 0, stream>>>(pooled, h1_w, h1_b, ln1_g, ln1_b,
                                 h2_w, h2_b, ln2_g, ln2_b, h3_w, h3_b, out);
}